// TDrumorGCN_15547781611783
// MI455X (gfx1250) — compile-verified
//
#include <hip/hip_runtime.h>
#include <stdint.h>

typedef float v2f __attribute__((ext_vector_type(2)));
typedef float v8f __attribute__((ext_vector_type(8)));

#define FDIM 64   // feature width of all per-node buffers (h, out1, out2)

// ----------------------------- small prep kernels -----------------------------

__global__ void k_init_dis(float* __restrict__ dis, int N) {
    int n = blockIdx.x * blockDim.x + threadIdx.x;
    if (n < N) dis[n] = 1.0f;              // appended self-loop contributes 1 to deg
}

__global__ void k_init_graph(int* __restrict__ root, int* __restrict__ cnt, int G) {
    int g = blockIdx.x * blockDim.x + threadIdx.x;
    if (g < G) { root[g] = 0x7fffffff; cnt[g] = 0; }
}

__global__ void k_deg_edges(const int* __restrict__ row, const int* __restrict__ col,
                            float* __restrict__ deg, int E) {
    int e = blockIdx.x * blockDim.x + threadIdx.x;
    if (e < E) {
        int r = row[e], c = col[e];
        if (r != c) unsafeAtomicAdd(&deg[r], 1.0f);   // ew=0 for original self-loops
    }
}

__global__ void k_root_cnt(const int* __restrict__ batch, int* __restrict__ root,
                           int* __restrict__ cnt, int N) {
    int n = blockIdx.x * blockDim.x + threadIdx.x;
    if (n < N) {
        int b = batch[n];
        atomicMin(&root[b], n);            // segment_min(arange, batch)
        atomicAdd(&cnt[b], 1);             // segment_sum(ones, batch)
    }
}

__global__ void k_finish_dis(float* __restrict__ dis, int N) {
    int n = blockIdx.x * blockDim.x + threadIdx.x;
    if (n < N) {
        float d = dis[n];
        dis[n] = (d > 0.0f) ? rsqrtf(d) : 0.0f;
    }
}

// ----------------------------- WMMA GEMM kernels ------------------------------
// H[nrows][64] = A[nrows][64] @ W[64][64] using V_WMMA_F32_16X16X4_F32.
// One wave per 16-row tile; 4 accumulators cover N=0..63; K loop step 4.

__global__ void k_gemm_xw(const float* __restrict__ A, const float* __restrict__ W,
                          float* __restrict__ H, int nrows) {
    const int lane = threadIdx.x & 31;
    const int wave = blockIdx.x * (blockDim.x >> 5) + (threadIdx.x >> 5);
    const int m0 = wave * 16;
    if (m0 >= nrows) return;               // wave-uniform: EXEC stays all-ones
    const int half = lane >> 4;            // 0: K={0,1}, 1: K={2,3}
    const int l15  = lane & 15;
    const int m    = m0 + l15;
    const int mc   = (m < nrows) ? m : (nrows - 1);
    const float* __restrict__ arow = A + (size_t)mc * FDIM;

    v8f c0 = {}; v8f c1 = {}; v8f c2 = {}; v8f c3 = {};
#pragma unroll
    for (int k = 0; k < 64; k += 4) {
        const int ka = k + half * 2;
        v2f a; a.x = arow[ka]; a.y = arow[ka + 1];
        const float* __restrict__ wk = W + (size_t)ka * 64;
        v2f b0; b0.x = wk[l15 +  0]; b0.y = wk[64 + l15 +  0];
        v2f b1; b1.x = wk[l15 + 16]; b1.y = wk[64 + l15 + 16];
        v2f b2; b2.x = wk[l15 + 32]; b2.y = wk[64 + l15 + 32];
        v2f b3; b3.x = wk[l15 + 48]; b3.y = wk[64 + l15 + 48];
        c0 = __builtin_amdgcn_wmma_f32_16x16x4_f32(false, a, false, b0, (short)0, c0, false, false);
        c1 = __builtin_amdgcn_wmma_f32_16x16x4_f32(false, a, false, b1, (short)0, c1, false, false);
        c2 = __builtin_amdgcn_wmma_f32_16x16x4_f32(false, a, false, b2, (short)0, c2, false, false);
        c3 = __builtin_amdgcn_wmma_f32_16x16x4_f32(false, a, false, b3, (short)0, c3, false, false);
    }
#pragma unroll
    for (int r = 0; r < 8; ++r) {
        const int mr = m0 + r + half * 8;
        if (mr < nrows) {
            float* __restrict__ hrow = H + (size_t)mr * FDIM;
            hrow[l15 +  0] = c0[r];
            hrow[l15 + 16] = c1[r];
            hrow[l15 + 32] = c2[r];
            hrow[l15 + 48] = c3[r];
        }
    }
}

// H[nrows][64] = relu([X2 | X[root(batch)]]) [nrows][128] @ W[128][64]
// Concat + ReLU fused into the A-operand load (saves a 51 MB intermediate).
__global__ void k_gemm_cat_w(const float* __restrict__ X2, const float* __restrict__ X,
                             const int* __restrict__ batch, const int* __restrict__ root,
                             const float* __restrict__ W, float* __restrict__ H, int nrows) {
    const int lane = threadIdx.x & 31;
    const int wave = blockIdx.x * (blockDim.x >> 5) + (threadIdx.x >> 5);
    const int m0 = wave * 16;
    if (m0 >= nrows) return;
    const int half = lane >> 4;
    const int l15  = lane & 15;
    const int m    = m0 + l15;
    const int mc   = (m < nrows) ? m : (nrows - 1);
    const float* __restrict__ arow0 = X2 + (size_t)mc * FDIM;
    const float* __restrict__ arow1 = X  + (size_t)root[batch[mc]] * FDIM;

    v8f c0 = {}; v8f c1 = {}; v8f c2 = {}; v8f c3 = {};
#pragma unroll
    for (int k = 0; k < 128; k += 4) {
        const int ka = k + half * 2;       // ka and ka+1 always on same side of 64
        float a0, a1;
        if (ka < 64) { a0 = arow0[ka];      a1 = arow0[ka + 1]; }
        else         { a0 = arow1[ka - 64]; a1 = arow1[ka - 63]; }
        v2f a; a.x = fmaxf(a0, 0.0f); a.y = fmaxf(a1, 0.0f);
        const float* __restrict__ wk = W + (size_t)ka * 64;
        v2f b0; b0.x = wk[l15 +  0]; b0.y = wk[64 + l15 +  0];
        v2f b1; b1.x = wk[l15 + 16]; b1.y = wk[64 + l15 + 16];
        v2f b2; b2.x = wk[l15 + 32]; b2.y = wk[64 + l15 + 32];
        v2f b3; b3.x = wk[l15 + 48]; b3.y = wk[64 + l15 + 48];
        c0 = __builtin_amdgcn_wmma_f32_16x16x4_f32(false, a, false, b0, (short)0, c0, false, false);
        c1 = __builtin_amdgcn_wmma_f32_16x16x4_f32(false, a, false, b1, (short)0, c1, false, false);
        c2 = __builtin_amdgcn_wmma_f32_16x16x4_f32(false, a, false, b2, (short)0, c2, false, false);
        c3 = __builtin_amdgcn_wmma_f32_16x16x4_f32(false, a, false, b3, (short)0, c3, false, false);
    }
#pragma unroll
    for (int r = 0; r < 8; ++r) {
        const int mr = m0 + r + half * 8;
        if (mr < nrows) {
            float* __restrict__ hrow = H + (size_t)mr * FDIM;
            hrow[l15 +  0] = c0[r];
            hrow[l15 + 16] = c1[r];
            hrow[l15 + 32] = c2[r];
            hrow[l15 + 48] = c3[r];
        }
    }
}

// ----------------------------- propagation kernels ----------------------------

// out[n][f] = dis[n]^2 * h[n][f] + bias[f]   (self-loop edge: norm = 1/deg)
__global__ void k_selfloop_bias(const float* __restrict__ h, const float* __restrict__ dis,
                                const float* __restrict__ bias, float* __restrict__ out, int N) {
    int idx = blockIdx.x * blockDim.x + threadIdx.x;
    if (idx < N * FDIM) {
        int n = idx >> 6, f = idx & 63;
        float w = dis[n] * dis[n];
        out[idx] = w * h[idx] + bias[f];
    }
}

// out[col] += dis[row]*dis[col] * h[row]   (one wave per edge; 2 features/lane)
__global__ void k_edge_scatter(const int* __restrict__ row, const int* __restrict__ col,
                               const float* __restrict__ dis, const float* __restrict__ h,
                               float* __restrict__ out, int E) {
    int lane = threadIdx.x & 31;
    int e = blockIdx.x * (blockDim.x >> 5) + (threadIdx.x >> 5);
    if (e >= E) return;
    int r = row[e], c = col[e];
    if (r == c) return;                    // ew == 0 for original self-loops
    float w = dis[r] * dis[c];
    const float* __restrict__ hr = h + (size_t)r * FDIM;
    float* __restrict__ oc = out + (size_t)c * FDIM;
    unsafeAtomicAdd(&oc[lane],      w * hr[lane]);
    unsafeAtomicAdd(&oc[lane + 32], w * hr[lane + 32]);
}

// dst[g][f<64]   = mean_n relu(out2[n][f])
// dst[g][f>=64]  = x2[root_g][f-64]  (mean of a per-graph constant)
__global__ void k_pool(const float* __restrict__ out2, const float* __restrict__ x2,
                       const int* __restrict__ root, const int* __restrict__ cnt,
                       float* __restrict__ dst) {
    int g = blockIdx.x;
    int f = threadIdx.x;                   // 0..127
    int start = root[g];
    int n = cnt[g];
    if (f < 64) {
        float s = 0.0f;
        for (int i = 0; i < n; ++i)
            s += fmaxf(out2[(size_t)(start + i) * FDIM + f], 0.0f);
        dst[(size_t)g * 128 + f] = s / (float)n;
    } else {
        dst[(size_t)g * 128 + f] = x2[(size_t)start * FDIM + (f - 64)];
    }
}

// --------------------------------- launcher -----------------------------------

extern "C" void kernel_launch(void* const* d_in, const int* in_sizes, int n_in,
                              void* d_out, int out_size, void* d_ws, size_t ws_size,
                              hipStream_t stream) {
    (void)n_in; (void)ws_size;
    const float* x     = (const float*)d_in[0];
    const int*   ei    = (const int*)d_in[1];
    const int*   batch = (const int*)d_in[2];
    const float* w1    = (const float*)d_in[3];
    const float* b1    = (const float*)d_in[4];
    const float* w2    = (const float*)d_in[5];
    const float* b2    = (const float*)d_in[6];
    float* out = (float*)d_out;

    const int N = in_sizes[0] / FDIM;      // 100000
    const int E = in_sizes[1] / 2;         // 1000000
    const int G = out_size / 128;          // 500
    const int* row = ei;
    const int* col = ei + E;

    // workspace layout (~77 MB)
    char* p = (char*)d_ws;
    float* dis  = (float*)p; p += (size_t)N * sizeof(float);
    int*   root = (int*)p;   p += (size_t)G * sizeof(int);
    int*   cnt  = (int*)p;   p += (size_t)G * sizeof(int);
    p = (char*)(((uintptr_t)p + 255) & ~(uintptr_t)255);
    float* hbuf = (float*)p; p += (size_t)N * FDIM * sizeof(float);  // h1, then h2
    float* out1 = (float*)p; p += (size_t)N * FDIM * sizeof(float);  // conv1 out (= x2)
    float* out2 = (float*)p;                                         // conv2 out

    const int nf = N * FDIM;
    const int tiles   = (N + 15) / 16;
    const int gblocks = (tiles + 7) / 8;   // 8 waves (16-row tiles) per 256-thread block

    // normalization + graph metadata
    k_init_dis  <<<(N + 255) / 256, 256, 0, stream>>>(dis, N);
    k_init_graph<<<(G + 255) / 256, 256, 0, stream>>>(root, cnt, G);
    k_deg_edges <<<(E + 255) / 256, 256, 0, stream>>>(row, col, dis, E);
    k_root_cnt  <<<(N + 255) / 256, 256, 0, stream>>>(batch, root, cnt, N);
    k_finish_dis<<<(N + 255) / 256, 256, 0, stream>>>(dis, N);

    // conv1: h1 = x @ w1 ; out1 = selfloop + scatter + b1
    k_gemm_xw      <<<gblocks, 256, 0, stream>>>(x, w1, hbuf, N);
    k_selfloop_bias<<<(nf + 255) / 256, 256, 0, stream>>>(hbuf, dis, b1, out1, N);
    k_edge_scatter <<<(E + 7) / 8, 256, 0, stream>>>(row, col, dis, hbuf, out1, E);

    // conv2: h2 = relu([out1 | x[root]]) @ w2 (fused) ; out2 = selfloop + scatter + b2
    k_gemm_cat_w   <<<gblocks, 256, 0, stream>>>(out1, x, batch, root, w2, hbuf, N);
    k_selfloop_bias<<<(nf + 255) / 256, 256, 0, stream>>>(hbuf, dis, b2, out2, N);
    k_edge_scatter <<<(E + 7) / 8, 256, 0, stream>>>(row, col, dis, hbuf, out2, E);

    // pooled output [G, 128]
    k_pool<<<G, 128, 0, stream>>>(out2, out1, root, cnt, out);
}